// CircularDND_57226144252001
// MI455X (gfx1250) — compile-verified
//
#include <hip/hip_runtime.h>
#include <stdint.h>

typedef __attribute__((ext_vector_type(2))) float v2f;
typedef __attribute__((ext_vector_type(8))) float v8f;

#define NKEY 128
#define QW   50
#define DELTA_F 1e-3f

// ---------------------------------------------------------------------------
// Kernel 0: initialize radix-select state + histogram (d_ws is poisoned).
// aux[0]=prefix_bits aux[1]=prefix_mask aux[2]=k_rem aux[3]=cnt_greater
// aux[4]=cnt_equal  aux[6]=total_sum(float)
// ---------------------------------------------------------------------------
__global__ void init_state(uint32_t* __restrict__ aux, uint32_t* __restrict__ hist) {
  int t = threadIdx.x;
  if (t < 256) hist[t] = 0u;
  if (t == 0) {
    aux[0] = 0u; aux[1] = 0u; aux[2] = QW; aux[3] = 0u; aux[4] = 0u; aux[5] = 0u;
  }
}

// ---------------------------------------------------------------------------
// Kernel 1: squared distances via V_WMMA_F32_16X16X4_F32.
// One wave handles a 16-row tile. X = keys_tile - key (in regs). Feeding the
// same registers as A and B computes D = X * X^T; the diagonal accumulates
// ||keys[m]-key||^2 over 32 K-chunks of 4. Also produces per-block partial
// sums of w = 1/(d+delta) for the global normalization term.
// ---------------------------------------------------------------------------
__global__ void dist_wmma(const float* __restrict__ key,
                          const float* __restrict__ keys,
                          float* __restrict__ w_arr,
                          float* __restrict__ partials,
                          int max_len, int ntiles) {
  __shared__ float wsum[8];
  const int wave = threadIdx.x >> 5;
  const int lane = threadIdx.x & 31;
  const int tile = blockIdx.x * 8 + wave;
  const int tclmp = (tile < ntiles) ? tile : (ntiles - 1);
  const int tbase = tclmp * 16;

  const int half = (lane < 16) ? 0 : 2;     // K sub-offset for this half-wave
  const int rloc = lane & 15;               // tile-local row
  int row  = tbase + rloc;
  int rowc = (row < max_len) ? row : (max_len - 1);

  const float* rp = keys + (size_t)rowc * NKEY + half;
  const float* kp = key + half;

  v8f c = {0.f, 0.f, 0.f, 0.f, 0.f, 0.f, 0.f, 0.f};
#pragma unroll
  for (int k = 0; k < NKEY; k += 4) {
    v2f a = *(const v2f*)(rp + k);          // keys[row][k+half .. k+half+1]
    v2f b = *(const v2f*)(kp + k);          // key [k+half .. k+half+1]
    v2f x = a - b;
    // D(16x16,f32) += X(16x4) * X^T(4x16): same regs as A and B.
    c = __builtin_amdgcn_wmma_f32_16x16x4_f32(
        false, x, false, x, (short)0, c, false, false);
  }

  // Diagonal element m lives in lane (m<8 ? m : m+16), accumulator VGPR m&7.
  float contrib = 0.f;
  const bool lo = (lane < 8);
  const bool hi = (lane >= 24);
  if (lo || hi) {
    int s = lo ? lane : (lane - 24);        // accumulator slot
    int m = lo ? lane : (lane - 16);        // tile-local row
    float d = c[0];
    d = (s == 1) ? c[1] : d;
    d = (s == 2) ? c[2] : d;
    d = (s == 3) ? c[3] : d;
    d = (s == 4) ? c[4] : d;
    d = (s == 5) ? c[5] : d;
    d = (s == 6) ? c[6] : d;
    d = (s == 7) ? c[7] : d;
    float w = 1.0f / (d + DELTA_F);
    int grow = tbase + m;
    if (tile < ntiles && grow < max_len) {
      w_arr[grow] = w;
      contrib = w;
    }
  }

  // Deterministic wave + block reduction of the weight sum.
#pragma unroll
  for (int off = 16; off > 0; off >>= 1) contrib += __shfl_xor(contrib, off);
  if (lane == 0) wsum[wave] = contrib;
  __syncthreads();
  if (threadIdx.x == 0) {
    float p = 0.f;
#pragma unroll
    for (int i = 0; i < 8; ++i) p += wsum[i];
    partials[blockIdx.x] = p;
  }
}

// ---------------------------------------------------------------------------
// Kernel 2: deterministic reduction of per-block partial sums.
// ---------------------------------------------------------------------------
__global__ void sum_partials(const float* __restrict__ partials, int n,
                             float* __restrict__ total) {
  __shared__ float lds[32];
  float acc = 0.f;
  for (int i = threadIdx.x; i < n; i += blockDim.x) acc += partials[i];
#pragma unroll
  for (int off = 16; off > 0; off >>= 1) acc += __shfl_xor(acc, off);
  int lane = threadIdx.x & 31, wave = threadIdx.x >> 5;
  if (lane == 0) lds[wave] = acc;
  __syncthreads();
  if (threadIdx.x == 0) {
    float t = 0.f;
    int nw = blockDim.x >> 5;
    for (int i = 0; i < nw; ++i) t += lds[i];
    *total = t;
  }
}

// ---------------------------------------------------------------------------
// Kernels 3/4 (x4 rounds): exact radix-select of the 50th-largest weight.
// Positive floats order-preserve as uint32 bit patterns.
// ---------------------------------------------------------------------------
__global__ void hist_pass(const float* __restrict__ w_arr, int n,
                          const uint32_t* __restrict__ aux,
                          uint32_t* __restrict__ hist, int shift) {
  __shared__ uint32_t lh[256];
  for (int i = threadIdx.x; i < 256; i += blockDim.x) lh[i] = 0u;
  __syncthreads();
  uint32_t pref = aux[0], mask = aux[1];
  int stride = gridDim.x * blockDim.x;
  for (int i = blockIdx.x * blockDim.x + threadIdx.x; i < n; i += stride) {
    uint32_t u = __float_as_uint(w_arr[i]);
    if ((u & mask) == pref) atomicAdd(&lh[(u >> shift) & 255u], 1u);
  }
  __syncthreads();
  for (int i = threadIdx.x; i < 256; i += blockDim.x)
    if (lh[i]) atomicAdd(&hist[i], lh[i]);
}

__global__ void select_digit(uint32_t* __restrict__ aux,
                             uint32_t* __restrict__ hist, int shift) {
  if (threadIdx.x == 0) {
    uint32_t k = aux[2];
    int chosen = 0;
    for (int d = 255; d >= 0; --d) {
      uint32_t h = hist[d];
      if (k <= h) { chosen = d; break; }
      k -= h;
    }
    aux[0] |= ((uint32_t)chosen) << shift;
    aux[1] |= 0xFFu << shift;
    aux[2] = k;                     // rank within the tie bucket
  }
  __syncthreads();
  for (int i = threadIdx.x; i < 256; i += blockDim.x) hist[i] = 0u;  // next round
}

// ---------------------------------------------------------------------------
// Kernel 5: gather indices of the top-QW weights.
// Exactly (QW - k_rem) elements are strictly > T; take k_rem ties == T.
// ---------------------------------------------------------------------------
__global__ void gather_topk(const float* __restrict__ w_arr, int n,
                            uint32_t* __restrict__ aux,
                            uint32_t* __restrict__ idx_buf) {
  uint32_t T = aux[0];
  uint32_t krem = aux[2];
  uint32_t gbase = QW - krem;
  int stride = gridDim.x * blockDim.x;
  for (int i = blockIdx.x * blockDim.x + threadIdx.x; i < n; i += stride) {
    uint32_t u = __float_as_uint(w_arr[i]);
    if (u > T) {
      uint32_t p = atomicAdd(&aux[3], 1u);
      if (p < 64u) idx_buf[p] = (uint32_t)i;
    } else if (u == T) {
      uint32_t p = atomicAdd(&aux[4], 1u);
      if (p < krem) {
        uint32_t s = gbase + p;
        if (s < 64u) idx_buf[s] = (uint32_t)i;
      }
    }
  }
}

// ---------------------------------------------------------------------------
// Kernel 6: out[c] = sum_j (w_j / total) * values[idx_j][c]   (128 channels)
// ---------------------------------------------------------------------------
__global__ void weighted_sum(const float* __restrict__ values,
                             const float* __restrict__ w_arr,
                             const uint32_t* __restrict__ idx_buf,
                             const float* __restrict__ total,
                             float* __restrict__ out) {
  int c = threadIdx.x;                       // 0..127
  float inv = 1.0f / (*total);
  float acc = 0.f;
  for (int j = 0; j < QW; ++j) {
    uint32_t idx = idx_buf[j];
    acc += w_arr[idx] * values[(size_t)idx * NKEY + c];
  }
  out[c] = acc * inv;
}

// ---------------------------------------------------------------------------
extern "C" void kernel_launch(void* const* d_in, const int* in_sizes, int n_in,
                              void* d_out, int out_size, void* d_ws, size_t ws_size,
                              hipStream_t stream) {
  const float* key    = (const float*)d_in[0];
  const float* keys   = (const float*)d_in[1];
  const float* values = (const float*)d_in[2];
  float* out = (float*)d_out;

  const int max_len = in_sizes[1] / NKEY;
  const int ntiles  = (max_len + 15) / 16;
  const int gridA   = (ntiles + 7) / 8;      // 8 waves (tiles) per 256-thread block

  // Workspace layout: [ weights: max_len f32 | aux: 8 u32 | hist: 256 u32 |
  //                     idx_buf: 64 u32 | partials: gridA f32 ]
  float*    w_arr    = (float*)d_ws;
  uint32_t* aux      = (uint32_t*)(w_arr + max_len);
  uint32_t* hist     = aux + 8;
  uint32_t* idx_buf  = hist + 256;
  float*    partials = (float*)(idx_buf + 64);
  float*    total    = (float*)(aux + 6);

  init_state<<<1, 256, 0, stream>>>(aux, hist);
  dist_wmma<<<gridA, 256, 0, stream>>>(key, keys, w_arr, partials, max_len, ntiles);
  sum_partials<<<1, 1024, 0, stream>>>(partials, gridA, total);
  for (int r = 0; r < 4; ++r) {
    int shift = 24 - 8 * r;
    hist_pass<<<1024, 256, 0, stream>>>(w_arr, max_len, aux, hist, shift);
    select_digit<<<1, 256, 0, stream>>>(aux, hist, shift);
  }
  gather_topk<<<1024, 256, 0, stream>>>(w_arr, max_len, aux, idx_buf);
  weighted_sum<<<1, 128, 0, stream>>>(values, w_arr, idx_buf, total, out);
}